// VolumeRenderer_28355374088234
// MI455X (gfx1250) — compile-verified
//
#include <hip/hip_runtime.h>
#include <hip/hip_bf16.h>

// ---------------------------------------------------------------------------
// PlenOctree volume renderer for MI455X (gfx1250, wave32).
// Latency-bound gather workload: working set (33.5MB data + 1.2MB child) is
// L2-resident (192MB L2). Strategy: per-ray threads with early exit, top tree
// levels staged to LDS via gfx1250 async global->LDS DMA, b128 leaf loads,
// native v_exp/v_rcp transcendentals, speculative global_prefetch of the next
// step's level-4 child line.
// ---------------------------------------------------------------------------

#define DEPTH        6
#define DATA_DIM     28
#define MAX_STEPS    192
#define STEP_SIZE    0.001f
#define BG_BRIGHT    1.0f

// Levels 0..3 of a full octree: (8^4-1)/7 = 585 nodes, 8 ints each.
#define LDS_NODES    585
#define LDS_INTS     (LDS_NODES * 8)   // 4680 ints = 18720 bytes

// Exact pointee type the async-LDS builtin expects (per clang diagnostic):
// GCC-style 16-byte int vector, addrspace(1) source, addrspace(3) dest.
typedef int v4i __attribute__((vector_size(4 * sizeof(int))));
typedef __attribute__((address_space(1))) v4i* gas_v4i_ptr;
typedef __attribute__((address_space(3))) v4i* las_v4i_ptr;

__device__ __forceinline__ float fast_rcp(float x) {
  return __builtin_amdgcn_rcpf(x);
}

__device__ __forceinline__ void dda_unit(float cx, float cy, float cz,
                                         float ix, float iy, float iz,
                                         float& tmin, float& tmax) {
  float t1x = -cx * ix, t2x = t1x + ix;
  float t1y = -cy * iy, t2y = t1y + iy;
  float t1z = -cz * iz, t2z = t1z + iz;
  float lo = fmaxf(fmaxf(fminf(t1x, t2x), fminf(t1y, t2y)), fminf(t1z, t2z));
  float hi = fminf(fminf(fmaxf(t1x, t2x), fmaxf(t1y, t2y)), fmaxf(t1z, t2z));
  tmin = fmaxf(0.0f, lo);
  tmax = fminf(1.0e9f, hi);
}

// Cheap LDS-only replay of the top 4 levels: returns the level-4 node index
// so we can prefetch its child line for the next step.
__device__ __forceinline__ int lds_descend4(float px, float py, float pz,
                                            const int* s_child, int stage) {
  int node = 0;
  float cx = 0.f, cy = 0.f, cz = 0.f;
  float csz = 1.f, rcsz = 1.f;
#pragma unroll
  for (int l = 0; l < 4; ++l) {
    float rx = (px - cx) * rcsz;
    float ry = (py - cy) * rcsz;
    float rz = (pz - cz) * rcsz;
    int ix = (rx >= 0.5f) ? 1 : 0;
    int iy = (ry >= 0.5f) ? 1 : 0;
    int iz = (rz >= 0.5f) ? 1 : 0;
    csz *= 0.5f; rcsz *= 2.0f;
    cx += ix ? csz : 0.0f;
    cy += iy ? csz : 0.0f;
    cz += iz ? csz : 0.0f;
    int ci = node * 8 + (ix << 2) + (iy << 1) + iz;
    int ch = (ci < stage) ? s_child[ci] : 0;
    if (ch == 0) return node;
    node += ch;
  }
  return node;
}

__global__ __launch_bounds__(256)
void VolumeRenderer_28355374088234_kernel(const float* __restrict__ data,
                                          const int*   __restrict__ child,
                                          const float* __restrict__ origins,
                                          const float* __restrict__ dirs,
                                          const float* __restrict__ viewdirs,
                                          const float* __restrict__ offset,
                                          const float* __restrict__ invradius,
                                          float*       __restrict__ out,
                                          int nrays, int n_child_ints) {
  __shared__ int s_child[LDS_INTS];
  const int stage = (n_child_ints < LDS_INTS) ? n_child_ints : LDS_INTS;

  // ---- Stage top 4 tree levels into LDS (gfx1250 async DMA path) ----------
#if __has_builtin(__builtin_amdgcn_global_load_async_to_lds_b128)
  {
    int nvec = stage >> 2;
    for (int i = threadIdx.x; i < nvec; i += blockDim.x) {
      __builtin_amdgcn_global_load_async_to_lds_b128(
          (gas_v4i_ptr)(child + 4 * i), (las_v4i_ptr)(&s_child[4 * i]),
          /*offset=*/0, /*cpol=*/0);
    }
    for (int i = (nvec << 2) + threadIdx.x; i < stage; i += blockDim.x)
      s_child[i] = child[i];
#if __has_builtin(__builtin_amdgcn_s_wait_asynccnt)
    __builtin_amdgcn_s_wait_asynccnt(0);
#else
    asm volatile("s_wait_asynccnt 0" ::: "memory");
#endif
  }
#else
  for (int i = threadIdx.x; i < stage; i += blockDim.x)
    s_child[i] = child[i];
#endif
  __syncthreads();

  const int ray = blockIdx.x * blockDim.x + threadIdx.x;
  if (ray >= nrays) return;

  // ---- Per-ray setup ------------------------------------------------------
  const float invr = invradius[0];
  const float delta_scale = 1.0f / invr;

  float ogx = offset[0] + origins[3 * ray + 0] * invr;
  float ogy = offset[1] + origins[3 * ray + 1] * invr;
  float ogz = offset[2] + origins[3 * ray + 2] * invr;

  float dx = dirs[3 * ray + 0];
  float dy = dirs[3 * ray + 1];
  float dz = dirs[3 * ray + 2];
  float rn = __frsqrt_rn(dx * dx + dy * dy + dz * dz);
  dx *= rn; dy *= rn; dz *= rn;

  const float ivx = fast_rcp(dx + 1e-9f);
  const float ivy = fast_rcp(dy + 1e-9f);
  const float ivz = fast_rcp(dz + 1e-9f);

  float t, tmax_ray;
  dda_unit(ogx, ogy, ogz, ivx, ivy, ivz, t, tmax_ray);

  // SH order-2 basis from the (pre-normalized) view direction.
  const float vx = viewdirs[3 * ray + 0];
  const float vy = viewdirs[3 * ray + 1];
  const float vz = viewdirs[3 * ray + 2];
  const float sh0 = 0.28209479177387814f;
  const float sh1 = -0.4886025119029199f * vy;
  const float sh2 = 0.4886025119029199f * vz;
  const float sh3 = -0.4886025119029199f * vx;
  const float sh4 = 1.0925484305920792f * vx * vy;
  const float sh5 = -1.0925484305920792f * vy * vz;
  const float sh6 = 0.31539156525252005f * (2.0f * vz * vz - vx * vx - vy * vy);
  const float sh7 = -1.0925484305920792f * vx * vz;
  const float sh8 = 0.5462742152960396f * (vx * vx - vy * vy);

  float light = 1.0f;
  float acc0 = 0.0f, acc1 = 0.0f, acc2 = 0.0f;

  if (t < tmax_ray) {
    for (int step = 0; step < MAX_STEPS; ++step) {
      const float px = ogx + t * dx;
      const float py = ogy + t * dy;
      const float pz = ogz + t * dz;

      // ---- Octree descent: levels 0..3 from LDS, deeper from global -------
      int node = 0, cell = 0;
      float cx = 0.f, cy = 0.f, cz = 0.f;
      float csz = 1.f, rcsz = 1.f;
      bool found = false;
#pragma unroll
      for (int lvl = 0; lvl < DEPTH; ++lvl) {
        float rx = (px - cx) * rcsz;
        float ry = (py - cy) * rcsz;
        float rz = (pz - cz) * rcsz;
        int ix = (rx >= 0.5f) ? 1 : 0;
        int iy = (ry >= 0.5f) ? 1 : 0;
        int iz = (rz >= 0.5f) ? 1 : 0;
        csz *= 0.5f; rcsz *= 2.0f;
        cx += ix ? csz : 0.0f;
        cy += iy ? csz : 0.0f;
        cz += iz ? csz : 0.0f;
        cell = (ix << 2) | (iy << 1) | iz;
        int ci = node * 8 + cell;
        int ch = (ci < stage) ? s_child[ci] : child[ci];
        if (ch == 0) { found = true; break; }
        node += ch;
      }
      if (!found) { node = 0; cell = 0; }  // reference fallback semantics

      // ---- Leaf payload: 28 floats, 112B, 16B-aligned -> 7x b128 ----------
      const float4* dp = reinterpret_cast<const float4*>(
          data + (size_t)(node * 8 + cell) * DATA_DIM);
      float4 q0 = dp[0], q1 = dp[1], q2 = dp[2], q3 = dp[3];
      float4 q4 = dp[4], q5 = dp[5], q6 = dp[6];

      // Independent DDA math overlaps the payload load latency.
      float ptx = (px - cx) * rcsz;
      float pty = (py - cy) * rcsz;
      float ptz = (pz - cz) * rcsz;
      float tmn, tmx;
      dda_unit(ptx, pty, ptz, ivx, ivy, ivz, tmn, tmx);
      const float dt = (tmx - tmn) * csz + STEP_SIZE;

      const float sigma = fmaxf(q6.w, 0.0f);                  // element 27
      const float att = __expf(-dt * sigma * delta_scale);    // v_exp_f32
      const float w = light * (1.0f - att);

      // SH dot products; coef layout is [c*9 + k] over elements 0..26.
      float s0 = sh0 * q0.x + sh1 * q0.y + sh2 * q0.z + sh3 * q0.w +
                 sh4 * q1.x + sh5 * q1.y + sh6 * q1.z + sh7 * q1.w + sh8 * q2.x;
      float s1 = sh0 * q2.y + sh1 * q2.z + sh2 * q2.w + sh3 * q3.x +
                 sh4 * q3.y + sh5 * q3.z + sh6 * q3.w + sh7 * q4.x + sh8 * q4.y;
      float s2 = sh0 * q4.z + sh1 * q4.w + sh2 * q5.x + sh3 * q5.y +
                 sh4 * q5.z + sh5 * q5.w + sh6 * q6.x + sh7 * q6.y + sh8 * q6.z;

      acc0 += w * fast_rcp(1.0f + __expf(-s0));
      acc1 += w * fast_rcp(1.0f + __expf(-s1));
      acc2 += w * fast_rcp(1.0f + __expf(-s2));

      light *= att;
      t += dt;
      if (t >= tmax_ray) break;

      // ---- Speculative prefetch of next step's level-4 child line ---------
      // (global_prefetch_b8): replay top-4 descent from LDS only, then warm
      // the first *global* link of the next pointer chain.
      {
        const float nx = ogx + t * dx;
        const float ny = ogy + t * dy;
        const float nz = ogz + t * dz;
        int pn = lds_descend4(nx, ny, nz, s_child, stage);
        __builtin_prefetch(child + pn * 8, 0, 0);
      }
    }
  }

  out[3 * ray + 0] = acc0 + BG_BRIGHT * light;
  out[3 * ray + 1] = acc1 + BG_BRIGHT * light;
  out[3 * ray + 2] = acc2 + BG_BRIGHT * light;
}

extern "C" void kernel_launch(void* const* d_in, const int* in_sizes, int n_in,
                              void* d_out, int out_size, void* d_ws, size_t ws_size,
                              hipStream_t stream) {
  const float* data      = (const float*)d_in[0];
  const int*   child     = (const int*)  d_in[1];
  const float* origins   = (const float*)d_in[2];
  const float* dirs      = (const float*)d_in[3];
  const float* viewdirs  = (const float*)d_in[4];
  const float* offset    = (const float*)d_in[5];
  const float* invradius = (const float*)d_in[6];

  const int nrays = in_sizes[2] / 3;
  const int n_child_ints = in_sizes[1];

  dim3 block(256);
  dim3 grid((nrays + 255) / 256);
  hipLaunchKernelGGL(VolumeRenderer_28355374088234_kernel, grid, block, 0, stream,
                     data, child, origins, dirs, viewdirs, offset, invradius,
                     (float*)d_out, nrays, n_child_ints);
}